// FFNPredictor_13864154432459
// MI455X (gfx1250) — compile-verified
//
#include <hip/hip_runtime.h>
#include <hip/hip_bf16.h>
#include <math.h>

// FFN channel predictor for MI455X (gfx1250, wave32).
// Pass 0: pack w1 into WMMA B-fragment order (256 KB, stays L2-resident).
// Pass 1: P[8192][16] = x @ [w1_b0|w1_b1] via V_WMMA_F32_16X16X4_F32, with x tiles
//         staged through LDS using GLOBAL_LOAD_ASYNC_TO_LDS_B64 (ASYNCcnt pipeline):
//         coalesced global reads, fragment-order LDS scatter, conflict-free ds_load.
// Pass 2: per-channel sum over tokens of (silu(q0)*q1)^2, fused WMMA, no q materialization.
// Pass 3: exact sorted top-5504 via bitonic sort of 16384 u64 keys in 128 KB LDS.

typedef __attribute__((ext_vector_type(2))) float v2f;
typedef __attribute__((ext_vector_type(8))) float v8f;

#define HDIM 4096
#define NTOK 8192
#define RDIM 8
#define IDIM 11008
#define KSEL 5504
#define SORTN 16384

#define CHUNK 256                          // x columns staged per step
#define NCHUNK (HDIM / CHUNK)              // 16
#define FRAGS_PER_CHUNK (CHUNK / 4)        // 64 WMMA K-slices per chunk
#define BUF_FLOATS (FRAGS_PER_CHUNK * 64)  // 4096 floats = 16 KB per buffer
#define NFRAG (HDIM / 4)                   // 1024 B-fragments total

static __device__ __forceinline__ v8f wmma_f32_4(v2f a, v2f b, v8f c) {
  return __builtin_amdgcn_wmma_f32_16x16x4_f32(false, a, false, b, (short)0, c,
                                               false, false);
}

// ---------------- Pass 0: pack w1 into B-fragment order ----------------------------
// Bp[f][lane] = { W[f*4 + (lane>>4)*2][lane&15], W[.. +1][lane&15] },
// W[h][c] = w1[c>>3][h][c&7].  One u64 per (frag,lane): 1024*32*8B = 256 KB.
__global__ __launch_bounds__(256) void ffn_pack_w1(const float* __restrict__ w1,
                                                   float* __restrict__ Bp) {
  const int t = blockIdx.x * 256 + threadIdx.x;  // 32768 threads
  if (t >= NFRAG * 32) return;
  const int f = t >> 5, l = t & 31;
  const int c = l & 15, half = l >> 4;
  const int g = c >> 3, r = c & 7;
  const int ka = f * 4 + half * 2;
  v2f b;
  b.x = w1[(size_t)g * HDIM * RDIM + (size_t)(ka + 0) * RDIM + r];
  b.y = w1[(size_t)g * HDIM * RDIM + (size_t)(ka + 1) * RDIM + r];
  *(v2f*)(Bp + (size_t)t * 2) = b;
}

// ---------------- Pass 1: P = x @ W with async-LDS staged A tiles ------------------
// grid = NTOK/16 blocks, 256 threads (8 waves). Block owns one 16-token tile.
// Wave w accumulates K-slices w*8..w*8+7 of each chunk; cross-wave LDS reduce at end.
__global__ __launch_bounds__(256) void ffn_pass1(const float* __restrict__ x,
                                                 const float* __restrict__ Bp,
                                                 float* __restrict__ P) {
  __shared__ float abuf[2][BUF_FLOATS];   // 32 KB double buffer, A-fragment order
  __shared__ float red[8][32][8];
  const int tile = blockIdx.x;
  const int t = threadIdx.x;
  const int wave = t >> 5, lane = t & 31;
  const int half = lane >> 4, l16 = lane & 15;

  // Per-thread staging plan: 2048 b64 transfers per 16x256 chunk -> 8 per thread.
  // Global side: thread t+j*256 reads pair 'pc' of row 'row' (coalesced 256B/wave).
  // LDS side: scatter into fragment order so consumers ds_load_b64 linearly.
  uint64_t gaddr[8];
  uint32_t laddr[8];
#pragma unroll
  for (int j = 0; j < 8; ++j) {
    const int p = t + j * 256;  // 0..2047
    const int row = p >> 7;     // 0..15
    const int pc = p & 127;     // pair-column within chunk
    gaddr[j] = (uint64_t)(uintptr_t)(x + (size_t)(tile * 16 + row) * HDIM + pc * 2);
    // fragment = pc>>1, lane-in-frag = (pc&1)*16 + row  -> float offset
    laddr[j] = (uint32_t)(uintptr_t)(&abuf[0][0]) +
               (uint32_t)((((pc >> 1) * 64) + (pc & 1) * 32 + row * 2) * 4);
  }

  auto stage = [&](int c, int b) {
    const uint64_t goff = (uint64_t)c * (CHUNK * 4);
    const uint32_t loff = (uint32_t)b * (BUF_FLOATS * 4);
#pragma unroll
    for (int j = 0; j < 8; ++j) {
      asm volatile("global_load_async_to_lds_b64 %0, %1, off"
                   :: "v"(laddr[j] + loff), "v"(gaddr[j] + goff)
                   : "memory");
    }
  };

  stage(0, 0);
  asm volatile("s_wait_asynccnt 0x0" ::: "memory");
  __syncthreads();

  v8f acc0 = {}, acc1 = {};
  for (int c = 0; c < NCHUNK; ++c) {
    const int b = c & 1;
    if (c + 1 < NCHUNK) stage(c + 1, b ^ 1);          // overlap next-chunk DMA
    const float* fb = &abuf[b][0];
    const float* bp = Bp + ((size_t)c * FRAGS_PER_CHUNK + wave * 8) * 64 + lane * 2;
#pragma unroll
    for (int jj = 0; jj < 8; jj += 2) {
      v2f a0 = *(const v2f*)(fb + (wave * 8 + jj + 0) * 64 + lane * 2);  // ds_load_b64
      v2f a1 = *(const v2f*)(fb + (wave * 8 + jj + 1) * 64 + lane * 2);
      v2f b0 = *(const v2f*)(bp + (jj + 0) * 64);                        // L2-resident
      v2f b1 = *(const v2f*)(bp + (jj + 1) * 64);
      acc0 = wmma_f32_4(a0, b0, acc0);
      acc1 = wmma_f32_4(a1, b1, acc1);
    }
    asm volatile("s_wait_asynccnt 0x0" ::: "memory");  // chunk c+1 resident
    __syncthreads();                                   // all waves done with buf b
  }

#pragma unroll
  for (int j = 0; j < 8; ++j) red[wave][lane][j] = acc0[j] + acc1[j];
  __syncthreads();
  if (wave == 0) {
    v8f s = {};
    for (int w = 0; w < 8; ++w)
#pragma unroll
      for (int j = 0; j < 8; ++j) s[j] += red[w][lane][j];
    // D layout: VGPR j holds (M = half*8 + j, col = l16)
#pragma unroll
    for (int j = 0; j < 8; ++j) {
      const int m = half * 8 + j;
      P[(size_t)(tile * 16 + m) * 16 + l16] = s[j];
    }
  }
}

// ---------------- Pass 2: nsq[i] = sum_n (silu(q0)*q1)^2 ---------------------------
// grid = IDIM/16 blocks, 256 threads (8 waves). Each block owns one 16-channel tile.
__global__ __launch_bounds__(256) void ffn_pass2(const float* __restrict__ P,
                                                 const float* __restrict__ w2,
                                                 float* __restrict__ nsq) {
  __shared__ float red[8][16];
  const int itile = blockIdx.x;
  const int wave = threadIdx.x >> 5;
  const int lane = threadIdx.x & 31;
  const int half = lane >> 4;
  const int l16  = lane & 15;
  const int ich  = itile * 16 + l16;

  const float* w20 = w2;
  const float* w21 = w2 + (size_t)RDIM * IDIM;
  const int kA = half * 2;
  v2f b00, b01, b10, b11;                 // hoisted B fragments (K=8 -> 2 WMMAs/branch)
  b00.x = w20[(size_t)(kA + 0) * IDIM + ich];
  b00.y = w20[(size_t)(kA + 1) * IDIM + ich];
  b01.x = w20[(size_t)(kA + 4) * IDIM + ich];
  b01.y = w20[(size_t)(kA + 5) * IDIM + ich];
  b10.x = w21[(size_t)(kA + 0) * IDIM + ich];
  b10.y = w21[(size_t)(kA + 1) * IDIM + ich];
  b11.x = w21[(size_t)(kA + 4) * IDIM + ich];
  b11.y = w21[(size_t)(kA + 5) * IDIM + ich];

  float sumsq = 0.f;
  for (int nt = wave; nt < NTOK / 16; nt += 8) {
    const float* prow = P + (size_t)(nt * 16 + l16) * 16;  // A row = token
    v2f a00 = *(const v2f*)(prow + kA);
    v2f a01 = *(const v2f*)(prow + 4 + kA);
    v2f a10 = *(const v2f*)(prow + 8 + kA);
    v2f a11 = *(const v2f*)(prow + 12 + kA);
    v8f q0 = {}; v8f q1 = {};
    q0 = wmma_f32_4(a00, b00, q0);
    q1 = wmma_f32_4(a10, b10, q1);
    q0 = wmma_f32_4(a01, b01, q0);
    q1 = wmma_f32_4(a11, b11, q1);
#pragma unroll
    for (int j = 0; j < 8; ++j) {
      const float z = q0[j];
      const float s = z * __frcp_rn(1.f + __expf(-z));  // silu
      const float g = s * q1[j];
      sumsq = fmaf(g, g, sumsq);
    }
  }
  sumsq += __shfl_xor(sumsq, 16, 32);  // fold lanes L and L+16 (same channel)
  if (half == 0) red[wave][l16] = sumsq;
  __syncthreads();
  if (threadIdx.x < 16) {
    float tacc = 0.f;
    for (int w = 0; w < 8; ++w) tacc += red[w][threadIdx.x];
    nsq[itile * 16 + threadIdx.x] = tacc;  // sqrt omitted: monotone for top-k
  }
}

// ---------------- Pass 3: sorted top-k via bitonic sort in LDS ---------------------
__global__ __launch_bounds__(1024) void ffn_topk(const float* __restrict__ nsq,
                                                 int* __restrict__ out) {
  extern __shared__ unsigned long long smem[];
  for (int i = threadIdx.x; i < SORTN; i += 1024) {
    unsigned long long v = 0ull;
    if (i < IDIM) {
      const unsigned int kb = __float_as_uint(nsq[i]);  // keys >= 0: bits monotone
      v = ((unsigned long long)kb << 32) | (unsigned int)(~i);  // ties -> lower idx
    }
    smem[i] = v;
  }
  __syncthreads();
  for (int size = 2; size <= SORTN; size <<= 1) {
    for (int stride = size >> 1; stride > 0; stride >>= 1) {
      for (int t = threadIdx.x; t < SORTN / 2; t += 1024) {
        const int lo = 2 * (t & ~(stride - 1)) + (t & (stride - 1));
        const int hi = lo + stride;
        const bool desc = ((lo & size) == 0);
        const unsigned long long a = smem[lo];
        const unsigned long long b = smem[hi];
        const bool doswap = desc ? (a < b) : (a > b);
        if (doswap) { smem[lo] = b; smem[hi] = a; }
      }
      __syncthreads();
    }
  }
  for (int i = threadIdx.x; i < KSEL; i += 1024) {
    out[i] = (int)(~(unsigned int)(smem[i] & 0xFFFFFFFFull));
  }
}

extern "C" void kernel_launch(void* const* d_in, const int* in_sizes, int n_in,
                              void* d_out, int out_size, void* d_ws, size_t ws_size,
                              hipStream_t stream) {
  (void)in_sizes; (void)n_in; (void)out_size; (void)ws_size;
  const float* x  = (const float*)d_in[0];   // [4,2048,4096] f32
  const float* w1 = (const float*)d_in[1];   // [2,4096,8]    f32
  const float* w2 = (const float*)d_in[2];   // [2,8,11008]   f32
  // d_in[3] = sparsity_pct (50) -> k = 5504, fixed by problem shape

  float* P   = (float*)d_ws;                                        // 512 KB
  float* Bp  = (float*)((char*)d_ws + (size_t)NTOK * 16 * 4);       // 256 KB
  float* nsq = (float*)((char*)d_ws + (size_t)NTOK * 16 * 4 + (size_t)NFRAG * 32 * 8);
  int*   out = (int*)d_out;                                         // 5504 indices

  ffn_pack_w1<<<(NFRAG * 32) / 256, 256, 0, stream>>>(w1, Bp);
  ffn_pass1<<<NTOK / 16, 256, 0, stream>>>(x, Bp, P);
  ffn_pass2<<<IDIM / 16, 256, 0, stream>>>(P, w2, nsq);
  ffn_topk<<<1, 1024, SORTN * sizeof(unsigned long long), stream>>>(nsq, out);
}